// FaultPredictionModel_84018150244815
// MI455X (gfx1250) — compile-verified
//
#include <hip/hip_runtime.h>
#include <math.h>

typedef float v2f __attribute__((ext_vector_type(2)));
typedef float v8f __attribute__((ext_vector_type(8)));

#define LROW 65536
#define BATCH 64
#define NCH 32
#define BLOCKS_PER_ROW 8
#define THREADS 256            // 8 wave32
#define WAVES 8
#define TILES_PER_ROW (LROW / 16)             // 4096 tiles of 16 positions
#define TILE_STRIDE (BLOCKS_PER_ROW * WAVES)  // 64
#define PARTIAL_STRIDE 40      // 32 deep-max + sum/sumsq/max/min, padded

// ---------------------------------------------------------------------------
// One 16-position tile: build the A fragment, run two f32 WMMAs (channels
// 0-15 and 16-31), max-accumulate raw conv outputs, accumulate stats.
// A layout (ISA 7.12.2, 32-bit A 16x4): lanes 0-15 hold K=0,1 in VGPR0/1,
// lanes 16-31 hold K=2,3. Branchless: neighbor index wraps inside the row
// (pa & (LROW-1)) and is zeroed via one cndmask when the wrap fired.
// ---------------------------------------------------------------------------
__device__ __forceinline__ void tile_step(
    const float* __restrict__ xb, int pos, int noff, int hi,
    v2f b0, v2f b1, v8f& acc0, v8f& acc1,
    float& s_sum, float& s_sq, float& s_max, float& s_min)
{
    const int pa  = pos + noff;
    const int paw = pa & (LROW - 1);
    float ax = xb[paw];
    const float xc = xb[pos];
    ax = (pa == paw) ? ax : 0.0f;

    v2f a;
    a.x = ax;
    a.y = hi ? 0.0f : xc;

    // prefetch the tile this wave touches 2 unrolled iterations ahead
    __builtin_prefetch(&xb[pos + 2 * TILE_STRIDE * 16], 0, 1);

    const v8f zero = {};
    v8f d0 = __builtin_amdgcn_wmma_f32_16x16x4_f32(
        false, a, false, b0, (short)0, zero, false, false);
    v8f d1 = __builtin_amdgcn_wmma_f32_16x16x4_f32(
        false, a, false, b1, (short)0, zero, false, false);

#pragma unroll
    for (int i = 0; i < 8; ++i) {
        acc0[i] = fmaxf(acc0[i], d0[i]);
        acc1[i] = fmaxf(acc1[i], d1[i]);
    }

    // all lanes accumulate (each element counted exactly twice -> halved
    // at reduction); max/min don't care about duplication
    s_sum += xc;
    s_sq  += xc * xc;
    s_max  = fmaxf(s_max, xc);
    s_min  = fminf(s_min, xc);
}

// ---------------------------------------------------------------------------
// Kernel 1: fused conv(width-3, 32ch) -> running max, + moment stats, one
// streaming pass over x. Conv taps done with V_WMMA_F32_16X16X4_F32:
//   A (16 pos x 4) = [x[p-1], x[p], x[p+1], 0] per row
//   B (4 x 16 ch)  = [w0; w1; w2; 0] per column
// ---------------------------------------------------------------------------
__global__ __launch_bounds__(THREADS)
void conv_stats_kernel(const float* __restrict__ x,
                       const float* __restrict__ conv_w,   // (32,1,3) flat
                       float* __restrict__ partials)
{
    const int row  = blockIdx.x / BLOCKS_PER_ROW;
    const int sub  = blockIdx.x % BLOCKS_PER_ROW;
    const float* __restrict__ xb = x + (size_t)row * LROW;

    const int lane = threadIdx.x & 31;
    const int wave = threadIdx.x >> 5;
    const int m    = lane & 15;      // matrix row (position) / B column (chan)
    const int hi   = lane >> 4;      // half-wave selector (K 0,1 vs K 2,3)
    const int noff = hi ? 1 : -1;    // neighbor offset this half-wave needs

    // --- B fragments: weights for channels 0-15 (b0) and 16-31 (b1) -------
    // lo half: VGPR0 = w[ch][0] (K0), VGPR1 = w[ch][1] (K1)
    // hi half: VGPR0 = w[ch][2] (K2), VGPR1 = 0       (K3)
    v2f b0, b1;
    {
        const int c0 = m, c1 = m + 16;
        b0.x = hi ? conv_w[c0 * 3 + 2] : conv_w[c0 * 3 + 0];
        b0.y = hi ? 0.0f               : conv_w[c0 * 3 + 1];
        b1.x = hi ? conv_w[c1 * 3 + 2] : conv_w[c1 * 3 + 0];
        b1.y = hi ? 0.0f               : conv_w[c1 * 3 + 1];
    }

    v8f acc0, acc1;                  // running max of raw conv outputs
#pragma unroll
    for (int i = 0; i < 8; ++i) { acc0[i] = -INFINITY; acc1[i] = -INFINITY; }

    float s_sum = 0.0f, s_sq = 0.0f;
    float s_max = -INFINITY, s_min = INFINITY;

    // Manual 2x unroll: 4096 tiles, start in [0,64), stride 128 -> exactly
    // 32 iterations per wave, second tile always in range (no remainder).
#pragma unroll 1
    for (int tile = sub * WAVES + wave; tile < TILES_PER_ROW;
         tile += 2 * TILE_STRIDE) {
        tile_step(xb, tile * 16 + m, noff, hi, b0, b1,
                  acc0, acc1, s_sum, s_sq, s_max, s_min);
        tile_step(xb, (tile + TILE_STRIDE) * 16 + m, noff, hi, b0, b1,
                  acc0, acc1, s_sum, s_sq, s_max, s_min);
    }

    // --- wave-level reduction ---------------------------------------------
    // C/D layout: lane holds channel N = lane&15 across all 8 VGPRs; lane and
    // lane+16 hold the same channel (different position rows).
    float cmax0 = -INFINITY, cmax1 = -INFINITY;
#pragma unroll
    for (int i = 0; i < 8; ++i) {
        cmax0 = fmaxf(cmax0, acc0[i]);
        cmax1 = fmaxf(cmax1, acc1[i]);
    }
    cmax0 = fmaxf(cmax0, __shfl_xor(cmax0, 16, 32));
    cmax1 = fmaxf(cmax1, __shfl_xor(cmax1, 16, 32));

#pragma unroll
    for (int off = 16; off >= 1; off >>= 1) {
        s_sum += __shfl_xor(s_sum, off, 32);
        s_sq  += __shfl_xor(s_sq,  off, 32);
        s_max  = fmaxf(s_max, __shfl_xor(s_max, off, 32));
        s_min  = fminf(s_min, __shfl_xor(s_min, off, 32));
    }

    // --- cross-wave reduction in LDS --------------------------------------
    __shared__ float lds_deep[WAVES][NCH];
    __shared__ float lds_sum[WAVES], lds_sq[WAVES], lds_max[WAVES], lds_min[WAVES];

    if (lane < 16) {
        lds_deep[wave][m]      = cmax0;
        lds_deep[wave][m + 16] = cmax1;
    }
    if (lane == 0) {
        lds_sum[wave] = s_sum * 0.5f;   // undo 2x duplication
        lds_sq[wave]  = s_sq * 0.5f;
        lds_max[wave] = s_max;
        lds_min[wave] = s_min;
    }
    __syncthreads();

    float* __restrict__ out = partials + (size_t)blockIdx.x * PARTIAL_STRIDE;
    if (threadIdx.x < NCH) {
        float v = -INFINITY;
#pragma unroll
        for (int w = 0; w < WAVES; ++w) v = fmaxf(v, lds_deep[w][threadIdx.x]);
        out[threadIdx.x] = v;
    } else if (threadIdx.x == NCH) {
        float a = 0.0f;
#pragma unroll
        for (int w = 0; w < WAVES; ++w) a += lds_sum[w];
        out[NCH] = a;
    } else if (threadIdx.x == NCH + 1) {
        float a = 0.0f;
#pragma unroll
        for (int w = 0; w < WAVES; ++w) a += lds_sq[w];
        out[NCH + 1] = a;
    } else if (threadIdx.x == NCH + 2) {
        float a = -INFINITY;
#pragma unroll
        for (int w = 0; w < WAVES; ++w) a = fmaxf(a, lds_max[w]);
        out[NCH + 2] = a;
    } else if (threadIdx.x == NCH + 3) {
        float a = INFINITY;
#pragma unroll
        for (int w = 0; w < WAVES; ++w) a = fminf(a, lds_min[w]);
        out[NCH + 3] = a;
    }
}

// ---------------------------------------------------------------------------
// Kernel 2: combine per-block partials, build h[36], run the tiny MLP head.
// One thread per batch row; ~5K MACs each — negligible.
// ---------------------------------------------------------------------------
__global__ __launch_bounds__(64)
void head_kernel(const float* __restrict__ partials,
                 const float* __restrict__ conv_b,
                 const float* __restrict__ w1, const float* __restrict__ b1,
                 const float* __restrict__ w2, const float* __restrict__ b2,
                 const float* __restrict__ w3, const float* __restrict__ b3,
                 const float* __restrict__ w4, const float* __restrict__ b4,
                 float* __restrict__ out)
{
    const int b = threadIdx.x;
    if (b >= BATCH) return;

    float deep[NCH];
#pragma unroll
    for (int i = 0; i < NCH; ++i) deep[i] = -INFINITY;
    float sum = 0.0f, sq = 0.0f, mx = -INFINITY, mn = INFINITY;

    for (int s = 0; s < BLOCKS_PER_ROW; ++s) {
        const float* p = partials + (size_t)(b * BLOCKS_PER_ROW + s) * PARTIAL_STRIDE;
#pragma unroll
        for (int i = 0; i < NCH; ++i) deep[i] = fmaxf(deep[i], p[i]);
        sum += p[NCH];
        sq  += p[NCH + 1];
        mx   = fmaxf(mx, p[NCH + 2]);
        mn   = fminf(mn, p[NCH + 3]);
    }

    float h[36];
#pragma unroll
    for (int i = 0; i < NCH; ++i) h[i] = fmaxf(deep[i] + conv_b[i], 0.0f);
    const float inv_l = 1.0f / (float)LROW;
    const float mean  = sum * inv_l;
    h[32] = mean;
    h[33] = sq * inv_l - mean * mean;   // E[x^2] - mean^2
    h[34] = mx;
    h[35] = mn;

    float h1[64];
    for (int j = 0; j < 64; ++j) {
        float a = b1[j];
#pragma unroll
        for (int i = 0; i < 36; ++i) a += h[i] * w1[i * 64 + j];
        h1[j] = fmaxf(a, 0.0f);
    }
    float h2[32];
    for (int j = 0; j < 32; ++j) {
        float a = b2[j];
#pragma unroll
        for (int i = 0; i < 64; ++i) a += h1[i] * w2[i * 32 + j];
        h2[j] = fmaxf(a, 0.0f);
    }
    float h3[16];
    for (int j = 0; j < 16; ++j) {
        float a = b3[j];
#pragma unroll
        for (int i = 0; i < 32; ++i) a += h2[i] * w3[i * 16 + j];
        h3[j] = fmaxf(a, 0.0f);
    }
#pragma unroll
    for (int j = 0; j < 2; ++j) {
        float a = b4[j];
#pragma unroll
        for (int i = 0; i < 16; ++i) a += h3[i] * w4[i * 2 + j];
        out[b * 2 + j] = a;
    }
}

extern "C" void kernel_launch(void* const* d_in, const int* in_sizes, int n_in,
                              void* d_out, int out_size, void* d_ws, size_t ws_size,
                              hipStream_t stream)
{
    const float* x      = (const float*)d_in[0];
    const float* conv_w = (const float*)d_in[1];
    const float* conv_b = (const float*)d_in[2];
    const float* w1 = (const float*)d_in[3];
    const float* b1 = (const float*)d_in[4];
    const float* w2 = (const float*)d_in[5];
    const float* b2 = (const float*)d_in[6];
    const float* w3 = (const float*)d_in[7];
    const float* b3 = (const float*)d_in[8];
    const float* w4 = (const float*)d_in[9];
    const float* b4 = (const float*)d_in[10];

    float* partials = (float*)d_ws;   // 512 blocks * 40 floats = 80 KB

    conv_stats_kernel<<<BATCH * BLOCKS_PER_ROW, THREADS, 0, stream>>>(
        x, conv_w, partials);
    head_kernel<<<1, 64, 0, stream>>>(
        partials, conv_b, w1, b1, w2, b2, w3, b3, w4, b4, (float*)d_out);
}